// HiPPOScale_29970281792126
// MI455X (gfx1250) — compile-verified
//
#include <hip/hip_runtime.h>

#define NDIM    256
#define BATCHSZ 32768
#define MT      128              // rows of c per workgroup
#define NT      64               // cols of out per workgroup
#define KC      32               // K chunk staged in LDS
#define NCHUNK  (NDIM / KC)      // 8
#define PAD     4                // keeps 16B row alignment + conflict-free banks
#define LDW     (KC + PAD)       // 36 floats per LDS row (144 B = 9*16)
#define THREADS 256              // 8 wave32

typedef __attribute__((ext_vector_type(2))) float v2f;
typedef __attribute__((ext_vector_type(8))) float v8f;
typedef __attribute__((ext_vector_type(4))) int   v4i;

typedef __attribute__((address_space(1))) v4i g_v4i;
typedef __attribute__((address_space(3))) v4i l_v4i;

#if defined(__HIP_DEVICE_COMPILE__) && __has_builtin(__builtin_amdgcn_global_load_async_to_lds_b128)
#define HAVE_ASYNC_LDS 1
#else
#define HAVE_ASYNC_LDS 0
#endif

// One 16-byte global -> LDS transfer per lane.
__device__ __forceinline__ void copy16_to_lds(const float* __restrict__ g, float* l) {
#if HAVE_ASYNC_LDS
    __builtin_amdgcn_global_load_async_to_lds_b128(
        (g_v4i*)(v4i*)g,
        (l_v4i*)(v4i*)l,
        /*imm offset*/0, /*cpol*/0);
#else
    const float4 v = *(const float4*)g;
    v2f lo; lo[0] = v.x; lo[1] = v.y;
    v2f hi; hi[0] = v.z; hi[1] = v.w;
    *(v2f*)l       = lo;
    *(v2f*)(l + 2) = hi;
#endif
}

__device__ __forceinline__ void wait_async_all() {
#if HAVE_ASYNC_LDS
#if __has_builtin(__builtin_amdgcn_s_wait_asynccnt)
    __builtin_amdgcn_s_wait_asynccnt(0);
#else
    asm volatile("s_wait_asynccnt 0x0" ::: "memory");
#endif
#endif
}

__global__ __launch_bounds__(THREADS)
void hippo_scale_wmma_f32(const float* __restrict__ c,
                          const float* __restrict__ f,
                          const float* __restrict__ Ast,
                          const float* __restrict__ Bst,
                          const int*   __restrict__ tptr,
                          float* __restrict__ out)
{
    __shared__ float lds_c[2][MT][LDW];
    __shared__ float lds_a[2][NT][LDW];
    __shared__ float lds_f[MT];

    const int t = tptr[0];
    const float* __restrict__ A = Ast + (size_t)t * NDIM * NDIM; // A[j*NDIM + k]
    const float* __restrict__ b = Bst + (size_t)t * NDIM;

    const int tid  = threadIdx.x;
    const int wave = tid >> 5;
    const int lane = tid & 31;
    const int lm   = lane & 15;   // M or N position within 16
    const int hl   = lane >> 4;   // half-wave: selects K pair / D row group
    const int wm   = wave & 3;    // wave M offset: wm*32
    const int wn   = wave >> 2;   // wave N offset: wn*32 (0..1)

    const int m_base = blockIdx.x * MT;
    const int n_base = blockIdx.y * NT;

    // Stage one KC-wide chunk of c (MT rows) and A (NT rows) into buffer `buf`.
    auto stage = [&](int buf, int kc) {
        // c tile: MT*KC/4 = 1024 b128 transfers -> 4 per thread
        #pragma unroll
        for (int j = 0; j < (MT * KC / 4) / THREADS; ++j) {
            const int idx = tid + j * THREADS;
            const int row = idx >> 3;      // / (KC/4)
            const int k4  = idx & 7;
            copy16_to_lds(&c[(size_t)(m_base + row) * NDIM + kc + k4 * 4],
                          &lds_c[buf][row][k4 * 4]);
        }
        // A tile: NT*KC/4 = 512 b128 transfers -> 2 per thread
        #pragma unroll
        for (int j = 0; j < (NT * KC / 4) / THREADS; ++j) {
            const int idx = tid + j * THREADS;
            const int row = idx >> 3;
            const int k4  = idx & 7;
            copy16_to_lds(&A[(size_t)(n_base + row) * NDIM + kc + k4 * 4],
                          &lds_a[buf][row][k4 * 4]);
        }
    };

    if (tid < MT) lds_f[tid] = f[m_base + tid];

    v8f acc00 = {}, acc01 = {}, acc10 = {}, acc11 = {};

    stage(0, 0);
    int buf = 0;

    for (int chunk = 0; chunk < NCHUNK; ++chunk) {
        wait_async_all();        // my current-chunk copies landed in LDS
        __syncthreads();         // everyone's copies landed; prev compute done
        if (chunk + 1 < NCHUNK)
            stage(buf ^ 1, (chunk + 1) * KC);   // overlap next loads w/ math

        #pragma unroll
        for (int k = 0; k < KC; k += 4) {
            const int kk = k + hl * 2;  // per-lane K pair per ISA frag layout
            const v2f a0 = *(const v2f*)&lds_c[buf][wm * 32 + lm     ][kk];
            const v2f a1 = *(const v2f*)&lds_c[buf][wm * 32 + 16 + lm][kk];
            const v2f b0 = *(const v2f*)&lds_a[buf][wn * 32 + lm     ][kk];
            const v2f b1 = *(const v2f*)&lds_a[buf][wn * 32 + 16 + lm][kk];
            acc00 = __builtin_amdgcn_wmma_f32_16x16x4_f32(false, a0, false, b0, (short)0, acc00, false, false);
            acc01 = __builtin_amdgcn_wmma_f32_16x16x4_f32(false, a0, false, b1, (short)0, acc01, false, false);
            acc10 = __builtin_amdgcn_wmma_f32_16x16x4_f32(false, a1, false, b0, (short)0, acc10, false, false);
            acc11 = __builtin_amdgcn_wmma_f32_16x16x4_f32(false, a1, false, b1, (short)0, acc11, false, false);
        }
        buf ^= 1;
    }

    // epilogue: out[m,n] = acc + b[n] * f[m]
    const int n0 = n_base + wn * 32 + lm;
    const int n1 = n0 + 16;
    const float bn0 = b[n0];
    const float bn1 = b[n1];
    #pragma unroll
    for (int r = 0; r < 8; ++r) {
        const int mloc0 = wm * 32 + hl * 8 + r;   // D layout: VGPR r -> M = 8*hl + r
        const int mloc1 = mloc0 + 16;
        const float f0 = lds_f[mloc0];
        const float f1 = lds_f[mloc1];
        out[(size_t)(m_base + mloc0) * NDIM + n0] = acc00[r] + bn0 * f0;
        out[(size_t)(m_base + mloc0) * NDIM + n1] = acc01[r] + bn1 * f0;
        out[(size_t)(m_base + mloc1) * NDIM + n0] = acc10[r] + bn0 * f1;
        out[(size_t)(m_base + mloc1) * NDIM + n1] = acc11[r] + bn1 * f1;
    }
}

extern "C" void kernel_launch(void* const* d_in, const int* in_sizes, int n_in,
                              void* d_out, int out_size, void* d_ws, size_t ws_size,
                              hipStream_t stream) {
    const float* c   = (const float*)d_in[0];
    const float* f   = (const float*)d_in[1];
    const float* Ast = (const float*)d_in[2];
    const float* Bst = (const float*)d_in[3];
    const int*   t   = (const int*)  d_in[4];
    float* out = (float*)d_out;

    dim3 grid(BATCHSZ / MT, NDIM / NT);   // (256, 4)
    hippo_scale_wmma_f32<<<grid, THREADS, 0, stream>>>(c, f, Ast, Bst, t, out);
}